// DenseVoxelPointNet_51659866636804
// MI455X (gfx1250) — compile-verified
//
#include <hip/hip_runtime.h>

// ---------------------------------------------------------------------------
// DenseVoxelPointNet for MI455X (gfx1250, wave32).
//
// Transposed formulation: one wave32 per voxel computes
//   h_tile(16x16) = W1^T(16x4, A) x feat^T(4x16, B)   [V_WMMA_F32_16X16X4_F32]
// so M = hidden, N = point. LayerNorm over hidden is then per-lane (8 VGPRs)
// plus ONE cross-half (xor16) combine; point reduction is one half16 reduce
// per VGPR; the 16x16 matvec needs only ONE cross-half add.
// All cross-lane movement uses VALU dpp8 / permlane16 / permlanex16
// (no ds_bpermute, no DScnt stalls).
// ---------------------------------------------------------------------------

typedef __attribute__((ext_vector_type(2))) float v2f;
typedef __attribute__((ext_vector_type(4))) float v4f;
typedef __attribute__((ext_vector_type(8))) float v8f;

#define LN_EPS 1e-5f

#if __has_builtin(__builtin_amdgcn_mov_dpp8) &&     \
    __has_builtin(__builtin_amdgcn_permlane16) &&   \
    __has_builtin(__builtin_amdgcn_permlanex16)
#define USE_VALU_XLANE 1
#else
#define USE_VALU_XLANE 0
#endif

#define DPP8SEL(a, b, c, d, e, f, g, h)                                      \
  ((a) | ((b) << 3) | ((c) << 6) | ((d) << 9) | ((e) << 12) | ((f) << 15) |  \
   ((g) << 18) | ((h) << 21))

#if USE_VALU_XLANE
template <int SEL>
__device__ __forceinline__ float fdpp8(float v) {
  return __int_as_float(__builtin_amdgcn_mov_dpp8(__float_as_int(v), SEL));
}
// lane l <- lane l^8 within each row of 16 (v_permlane16_b32)
__device__ __forceinline__ float fxor8(float v) {
  int s = __float_as_int(v);
  return __int_as_float(__builtin_amdgcn_permlane16(
      s, s, (int)0xFEDCBA98, (int)0x76543210, false, false));
}
// lane l <- lane l^16 (v_permlanex16_b32, identity select across rows)
__device__ __forceinline__ float fxor16(float v) {
  int s = __float_as_int(v);
  return __int_as_float(__builtin_amdgcn_permlanex16(
      s, s, (int)0x76543210, (int)0xFEDCBA98, false, false));
}
#endif

// Sum across the 16 lanes of this lane's half-wave (lanes 0-15 or 16-31).
__device__ __forceinline__ float half16_sum(float x) {
#if USE_VALU_XLANE
  x += fdpp8<DPP8SEL(1, 0, 3, 2, 5, 4, 7, 6)>(x);  // xor 1
  x += fdpp8<DPP8SEL(2, 3, 0, 1, 6, 7, 4, 5)>(x);  // xor 2
  x += fdpp8<DPP8SEL(4, 5, 6, 7, 0, 1, 2, 3)>(x);  // xor 4
  x += fxor8(x);                                   // xor 8
#else
  x += __shfl_xor(x, 1, 32);
  x += __shfl_xor(x, 2, 32);
  x += __shfl_xor(x, 4, 32);
  x += __shfl_xor(x, 8, 32);
#endif
  return x;
}

// Value of this lane's partner in the other half-wave (lane ^ 16).
__device__ __forceinline__ float other_half(float x) {
#if USE_VALU_XLANE
  return fxor16(x);
#else
  return __shfl_xor(x, 16, 32);
#endif
}

__global__ void __launch_bounds__(256) dvpn_zero_kernel(v4f* __restrict__ out4,
                                                        long n4) {
  long i = (long)blockIdx.x * blockDim.x + threadIdx.x;
  const long stride = (long)gridDim.x * blockDim.x;
  const v4f z = {0.f, 0.f, 0.f, 0.f};
  // Non-temporal: 512 MiB streaming fill, keep it out of L2.
  for (; i < n4; i += stride) __builtin_nontemporal_store(z, out4 + i);
}

__global__ void __launch_bounds__(256) dvpn_main_kernel(
    const float* __restrict__ feat,        // (V,32,4)
    const int*   __restrict__ num_points,  // (V,)
    const int*   __restrict__ coords,      // (V,4) = b,ix,iy,iz
    const float* __restrict__ W1,          // (4,16)
    const float* __restrict__ b1,
    const float* __restrict__ g1,
    const float* __restrict__ be1,
    const float* __restrict__ W2,          // (16,16)
    const float* __restrict__ b2,
    const float* __restrict__ g2,
    const float* __restrict__ be2,
    const int*   __restrict__ pGH,
    const int*   __restrict__ pGW,
    const int*   __restrict__ pGZ,
    float* __restrict__ out,               // (1,GH,GW,GZ,16)
    int V) {
  const int  lane = threadIdx.x & 31;
  const int  col  = lane & 15;  // point index within tile / output channel
  const bool hi   = lane >= 16;
  const int  waveInBlock   = threadIdx.x >> 5;
  const int  wavesPerBlock = blockDim.x >> 5;
  const int  firstWave = blockIdx.x * wavesPerBlock + waveInBlock;
  const int  nWaves    = gridDim.x * wavesPerBlock;

  const long GH = *pGH, GW = *pGW, GZ = *pGZ;

  // A = W1^T (16 hidden x 4 in) in WMMA f32 A layout:
  //   lane<16: M=col, K={0,1}; lane>=16: M=col, K={2,3}.
  v2f w1a;
  w1a.x = W1[(hi ? 2 : 0) * 16 + col];
  w1a.y = W1[(hi ? 3 : 1) * 16 + col];

  // This lane owns hidden rows [hbase, hbase+8) of the D tiles.
  const int hbase = hi ? 8 : 0;
  float b1v[8], g1v[8], be1v[8], w2v[8];
#pragma unroll
  for (int i = 0; i < 8; ++i) {
    b1v[i]  = b1[hbase + i];
    g1v[i]  = g1[hbase + i];
    be1v[i] = be1[hbase + i];
    w2v[i]  = W2[(hbase + i) * 16 + col];  // W2 rows owned by this half
  }
  const float b2n = b2[col], g2n = g2[col], be2n = be2[col];

  for (int v = firstWave; v < V; v += nWaves) {  // wave-uniform loop
    const float* fv = feat + (size_t)v * 128;

    // B tiles = feat^T (4 x 16 points), f32 B layout mirrors A layout:
    //   lane<16: K={0,1} of point col; lane>=16: K={2,3} of point col.
    const int koff = hi ? 2 : 0;
    v2f f0 = *(const v2f*)(fv + col * 4 + koff);         // points 0-15
    v2f f1 = *(const v2f*)(fv + (16 + col) * 4 + koff);  // points 16-31

    v8f c0 = {};
    v8f c1 = {};
    c0 = __builtin_amdgcn_wmma_f32_16x16x4_f32(false, w1a, false, f0,
                                               (short)0, c0, false, false);
    c1 = __builtin_amdgcn_wmma_f32_16x16x4_f32(false, w1a, false, f1,
                                               (short)0, c1, false, false);

    const int np = num_points[v];
    const bool m0 = col < np;        // point col valid (tile 0)
    const bool m1 = (16 + col) < np; // point 16+col valid (tile 1)

    float acc[8];
#pragma unroll
    for (int i = 0; i < 8; ++i) acc[i] = 0.f;

    // --- tile 0: +b1, LayerNorm over hidden(16), ReLU, masked accumulate ---
    {
      float val[8];
      float ps = 0.f;
#pragma unroll
      for (int i = 0; i < 8; ++i) {
        val[i] = c0[i] + b1v[i];
        ps += val[i];
      }
      const float mu = (ps + other_half(ps)) * 0.0625f;
      float pq = 0.f;
#pragma unroll
      for (int i = 0; i < 8; ++i) {
        val[i] -= mu;
        pq = fmaf(val[i], val[i], pq);
      }
      const float inv = rsqrtf((pq + other_half(pq)) * 0.0625f + LN_EPS);
#pragma unroll
      for (int i = 0; i < 8; ++i) {
        const float h = fmaxf(val[i] * inv * g1v[i] + be1v[i], 0.f);
        acc[i] += m0 ? h : 0.f;
      }
    }
    // --- tile 1 ---
    {
      float val[8];
      float ps = 0.f;
#pragma unroll
      for (int i = 0; i < 8; ++i) {
        val[i] = c1[i] + b1v[i];
        ps += val[i];
      }
      const float mu = (ps + other_half(ps)) * 0.0625f;
      float pq = 0.f;
#pragma unroll
      for (int i = 0; i < 8; ++i) {
        val[i] -= mu;
        pq = fmaf(val[i], val[i], pq);
      }
      const float inv = rsqrtf((pq + other_half(pq)) * 0.0625f + LN_EPS);
#pragma unroll
      for (int i = 0; i < 8; ++i) {
        const float h = fmaxf(val[i] * inv * g1v[i] + be1v[i], 0.f);
        acc[i] += m1 ? h : 0.f;
      }
    }

    // Masked sum over all 32 points (lanes of this half) + partial matvec:
    // lanes<16 contribute W2 rows 0-7, lanes>=16 rows 8-15.
    float xpart = 0.f;
#pragma unroll
    for (int i = 0; i < 8; ++i) {
      const float s = half16_sum(acc[i]);  // s[hbase+i], uniform per half
      xpart = fmaf(s, w2v[i], xpart);
    }
    const float x = xpart + other_half(xpart) + (float)np * b2n;

    // Second LayerNorm over the 16 outputs (x symmetric across halves).
    const float mu  = half16_sum(x) * 0.0625f;
    const float d   = x - mu;
    const float var = half16_sum(d * d) * 0.0625f;
    const float y   = d * rsqrtf(var + LN_EPS) * g2n + be2n;

    // Scatter: 16 contiguous floats at the voxel's grid cell.
    const int4 cc = ((const int4*)coords)[v];  // b, ix, iy, iz
    const long base =
        ((((long)cc.x * GH + cc.y) * GW + cc.z) * GZ + cc.w) * 16;
    if (!hi) out[base + col] = y;
  }
}

extern "C" void kernel_launch(void* const* d_in, const int* in_sizes, int n_in,
                              void* d_out, int out_size, void* d_ws,
                              size_t ws_size, hipStream_t stream) {
  const float* feat       = (const float*)d_in[0];
  const int*   num_points = (const int*)d_in[1];
  const int*   coords     = (const int*)d_in[2];
  const float* W1  = (const float*)d_in[3];
  const float* b1  = (const float*)d_in[4];
  const float* g1  = (const float*)d_in[5];
  const float* be1 = (const float*)d_in[6];
  const float* W2  = (const float*)d_in[7];
  const float* b2  = (const float*)d_in[8];
  const float* g2  = (const float*)d_in[9];
  const float* be2 = (const float*)d_in[10];
  // d_in[11] = batch_size (unused; coords carry b), [12..14] = GH, GW, GZ
  const int* pGH = (const int*)d_in[12];
  const int* pGW = (const int*)d_in[13];
  const int* pGZ = (const int*)d_in[14];

  float* out = (float*)d_out;
  const int V = in_sizes[1];  // num_points has V elements

  // 1) Zero-fill the dense grid (512 MiB of f32) — the dominant HBM cost.
  const long n4 = (long)out_size / 4;
  dvpn_zero_kernel<<<8192, 256, 0, stream>>>((v4f*)out, n4);

  // 2) PointNet + scatter: one wave32 per voxel, grid-stride.
  dvpn_main_kernel<<<2048, 256, 0, stream>>>(feat, num_points, coords, W1, b1,
                                             g1, be1, W2, b2, g2, be2, pGH,
                                             pGW, pGZ, out, V);
}